// GaussianMixture_163208757502
// MI455X (gfx1250) — compile-verified
//
#include <hip/hip_runtime.h>

typedef __attribute__((ext_vector_type(2))) float v2f;
typedef __attribute__((ext_vector_type(8))) float v8f;

#define DD  64   // feature dim D
#define ZLD 65   // z row pitch: [u | x(64)]

// ---------------------------------------------------------------------------
// Phase 1: idx[n] = clip(searchsorted_right(mix_partition, u_n), 0, K-1)
//          + per-cluster histogram.
// ---------------------------------------------------------------------------
__global__ void gm_phase1_idx_count(const float* __restrict__ z,
                                    const float* __restrict__ part,
                                    int* __restrict__ idx_arr,
                                    int* __restrict__ counts,
                                    int N, int K) {
  int n = blockIdx.x * blockDim.x + threadIdx.x;
  if (n >= N) return;
  float u = z[(long)n * ZLD];
  int lo = 0, hi = K;
  while (lo < hi) {                 // # of elements <= u  (side='right')
    int mid = (lo + hi) >> 1;
    if (part[mid] <= u) lo = mid + 1; else hi = mid;
  }
  int k = lo < (K - 1) ? lo : (K - 1);
  idx_arr[n] = k;
  atomicAdd(&counts[k], 1);
}

// ---------------------------------------------------------------------------
// Phase 2: exclusive scan of counts, each segment padded up to multiple of 16
//          so every 16-row GEMM tile lies inside one cluster.
// ---------------------------------------------------------------------------
__global__ void gm_phase2_scan(const int* __restrict__ counts,
                               int* __restrict__ po,
                               int* __restrict__ cursor, int K) {
  if (blockIdx.x == 0 && threadIdx.x == 0) {
    int acc = 0;
    for (int k = 0; k < K; ++k) {
      po[k] = acc;
      cursor[k] = acc;
      acc += (counts[k] + 15) & ~15;
    }
    po[K] = acc;
  }
}

// ---------------------------------------------------------------------------
// Phase 3: scatter row ids into padded, cluster-sorted permutation.
// Pad slots keep the -1 sentinel from the 0xFF memset.
// ---------------------------------------------------------------------------
__global__ void gm_phase3_scatter(const int* __restrict__ idx_arr,
                                  int* __restrict__ cursor,
                                  int* __restrict__ perm, int N) {
  int n = blockIdx.x * blockDim.x + threadIdx.x;
  if (n >= N) return;
  int k = idx_arr[n];
  int pos = atomicAdd(&cursor[k], 1);
  perm[pos] = n;
}

// ---------------------------------------------------------------------------
// Phase 4: per 16-row tile (one wave32 each):
//   Y(16x64) = X(16x64) * devs[k]^T + means[k], via V_WMMA_F32_16X16X4_F32.
//   4 column tiles x 16 K-steps = 64 WMMAs per wave, 100% useful FLOPs.
//
// Padded (perm == -1) rows: lanes load garbage from z row 0 UNCONDITIONALLY
// (safe address). WMMA row m of D only depends on row m of A, and padded
// rows are never stored, so no zeroing / predication is needed on the
// A-operand path -- keeps loads branch-free and clause-able.
//
// Wave32 operand layouts (ISA 7.12.2):
//   A 16x4 f32 (2 VGPRs): lanes 0-15 -> M=lane, K={j0,j0+1};
//                         lanes 16-31 -> M=lane-16, K={j0+2,j0+3}
//   B 4x16 f32 (2 VGPRs): lanes 0-15 -> N=lane, K={j0,j0+1};
//                         lanes 16-31 -> N=lane-16, K={j0+2,j0+3}
//   C/D 16x16 f32 (8 VGPRs): vgpr r: lanes 0-15 -> (M=r,   N=lane),
//                                    lanes 16-31 -> (M=r+8, N=lane-16)
// ---------------------------------------------------------------------------
__global__ void __launch_bounds__(128)
gm_phase4_wmma(const float* __restrict__ z,
               const float* __restrict__ means,
               const float* __restrict__ devs,
               const int*  __restrict__ po,
               const int*  __restrict__ perm,
               float* __restrict__ out, int K) {
  const int lane = threadIdx.x & 31;
  const int wave = threadIdx.x >> 5;
  const int tile = blockIdx.x * 4 + wave;
  const int base = tile * 16;

  const int total = po[K];
  if (base >= total) return;            // wave-uniform: whole wave exits

  // Locate cluster k with po[k] <= base < po[k+1] (binary search, uniform).
  int lo = 0, hi = K - 1;
  while (lo < hi) {
    int mid = (lo + hi + 1) >> 1;
    if (po[mid] <= base) lo = mid; else hi = mid - 1;
  }
  const int k = lo;

  const int hl  = lane >> 4;            // lane half: 0 or 1
  const int l16 = lane & 15;

  // Row this lane feeds into the A operand (M = l16). -1 => padded slot:
  // redirect to row 0 (safe garbage, discarded at store time).
  const int rowA = perm[base + l16];
  const float* xA = z + ((rowA >= 0) ? ((long)rowA * ZLD + 1) : 1L);
  const float* devk = devs + (long)k * DD * DD;

  v8f acc[4] = {v8f{}, v8f{}, v8f{}, v8f{}};

#pragma unroll
  for (int j0 = 0; j0 < DD; j0 += 4) {
    const int kk = j0 + 2 * hl;         // this lane's two K slots
    v2f a;
    a.x = xA[kk];                       // unconditional, immediate offsets
    a.y = xA[kk + 1];
#pragma unroll
    for (int c = 0; c < 4; ++c) {
      // B[j, i] = devs[k][i, j]; lane holds column i = c*16 + l16,
      // rows kk, kk+1 -> contiguous pair in devs[k] row i (8B aligned).
      const float* brow = devk + (long)(c * 16 + l16) * DD + kk;
      v2f b;
      b.x = brow[0];
      b.y = brow[1];
      acc[c] = __builtin_amdgcn_wmma_f32_16x16x4_f32(
          false, a, false, b, (short)0, acc[c], false, false);
    }
  }

  // Epilogue. acc[c][r] = D(M = r + 8*hl, N = l16) of column tile c.
  // r outer so the row-validity branch is taken once per tile row, with
  // 4 stores at immediate offsets from one row base inside it.
  float mval[4];
#pragma unroll
  for (int c = 0; c < 4; ++c) mval[c] = means[(long)k * DD + c * 16 + l16];

#pragma unroll
  for (int r = 0; r < 8; ++r) {
    const int m = hl * 8 + r;                 // tile row
    const int row = perm[base + m];
    if (row >= 0) {
      float* orow = out + (long)row * DD + l16;
#pragma unroll
      for (int c = 0; c < 4; ++c) {
        orow[c * 16] = acc[c][r] + mval[c];
      }
    }
  }
}

// ---------------------------------------------------------------------------
// Launch. Workspace (ints): counts[K] | po[K+1] | cursor[K] | idx[N] | perm[N+16K]
//   = 2N + 19K + 1 ints  (~529 KB for N=65536, K=64).
// ---------------------------------------------------------------------------
extern "C" void kernel_launch(void* const* d_in, const int* in_sizes, int n_in,
                              void* d_out, int out_size, void* d_ws, size_t ws_size,
                              hipStream_t stream) {
  (void)n_in; (void)out_size; (void)ws_size;
  const float* z     = (const float*)d_in[0];
  const float* means = (const float*)d_in[1];
  const float* devs  = (const float*)d_in[2];
  const float* part  = (const float*)d_in[3];
  float* out = (float*)d_out;

  const int K = in_sizes[3];
  const int N = in_sizes[0] / ZLD;
  const int padN = N + 16 * K;

  int* ws      = (int*)d_ws;
  int* counts  = ws;                 // K
  int* po      = counts + K;         // K + 1
  int* cursor  = po + K + 1;         // K
  int* idx_arr = cursor + K;         // N
  int* perm    = idx_arr + N;        // padN

  hipMemsetAsync(counts, 0, (size_t)K * sizeof(int), stream);
  hipMemsetAsync(perm, 0xFF, (size_t)padN * sizeof(int), stream);  // -1 sentinel

  gm_phase1_idx_count<<<(N + 255) / 256, 256, 0, stream>>>(z, part, idx_arr, counts, N, K);
  gm_phase2_scan<<<1, 1, 0, stream>>>(counts, po, cursor, K);
  gm_phase3_scatter<<<(N + 255) / 256, 256, 0, stream>>>(idx_arr, cursor, perm, N);

  const int tiles = padN / 16;       // upper bound; extra tiles exit early
  gm_phase4_wmma<<<(tiles + 3) / 4, 128, 0, stream>>>(z, means, devs, po, perm, out, K);
}